// GATWithEdgeAttr_49014166782221
// MI455X (gfx1250) — compile-verified
//
#include <hip/hip_runtime.h>
#include <math.h>

#define NN 100000
#define LL 48
#define EE 1600000
#define EDD 17
#define HH 48
#define MLP_WAVES 8
#define MLP_EPB (MLP_WAVES * 16)   // 128 edges per block

typedef __attribute__((ext_vector_type(2))) float v2f;
typedef __attribute__((ext_vector_type(8))) float v8f;

__device__ __forceinline__ float gelu_exact(float x) {
    return 0.5f * x * (1.0f + erff(x * 0.70710678118f));
}
__device__ __forceinline__ float sigmoidf_(float x) {
    return 1.0f / (1.0f + expf(-x));
}

// ---------------------------------------------------------------------------
// Kernel: init x = nodes*valid, v = valid, orig = x, agg = 0
// ---------------------------------------------------------------------------
__global__ void init_kernel(const float* __restrict__ nodes, const float* __restrict__ valid,
                            float* __restrict__ x, float* __restrict__ v,
                            float* __restrict__ orig, float* __restrict__ agg, int n) {
    int t = blockIdx.x * blockDim.x + threadIdx.x;
    if (t < n) {
        float vv = valid[t];
        float xx = nodes[t] * vv;
        x[t] = xx;
        orig[t] = xx;
        v[t] = vv;
        agg[t] = 0.0f;
    }
}

// ---------------------------------------------------------------------------
// Kernel: per-node row mean of v (48 cols). Half-wave (16 lanes) per node.
// ---------------------------------------------------------------------------
__global__ void vmean_kernel(const float* __restrict__ v, float* __restrict__ vmean, int n) {
    int t = blockIdx.x * blockDim.x + threadIdx.x;
    int i = t >> 4;
    int j = t & 15;
    if (i >= n) return;
    const float* vr = v + (size_t)i * LL;
    float s = vr[j] + vr[j + 16] + vr[j + 32];
    s += __shfl_xor(s, 1, 16);
    s += __shfl_xor(s, 2, 16);
    s += __shfl_xor(s, 4, 16);
    s += __shfl_xor(s, 8, 16);
    if (j == 0) vmean[i] = s * (1.0f / 48.0f);
}

// ---------------------------------------------------------------------------
// WMMA helper: acc(16x48) += A(16 x 4*KC from LDS) @ Wp (pair-swizzled LDS).
// A fragment (V_WMMA_F32_16X16X4_F32): lanes 0-15 M=lane K{k0,k0+1},
// lanes 16-31 M=lane-16 K{k0+2,k0+3}; B mirrors with N=lane&15.
// Pair-swizzled weights: Wp[(k>>1)*96 + col*2 + (k&1)] -> every fragment
// element pair is one aligned 8-byte ds_load_b64, no guards needed.
// ---------------------------------------------------------------------------
template <int KC, int STRIDE>
__device__ __forceinline__ void gemm_tile_lds(const float* __restrict__ Asrc,
                                              const float* __restrict__ Wp,
                                              int row, int khalf, v8f acc[3]) {
#pragma unroll
    for (int kc = 0; kc < KC; ++kc) {
        const int kp = kc * 2 + khalf;                 // K-pair index (k0 = 2*kp)
        v2f a = *(const v2f*)(Asrc + row * STRIDE + kp * 2);
#pragma unroll
        for (int nb = 0; nb < 3; ++nb) {
            v2f b = *(const v2f*)(Wp + kp * 96 + (nb * 16 + row) * 2);
            acc[nb] = __builtin_amdgcn_wmma_f32_16x16x4_f32(
                false, a, false, b, (short)0, acc[nb], false, false);
        }
    }
}

// ---------------------------------------------------------------------------
// Kernel: loop-invariant edge MLP -> per-edge gate scalar w_e.
// 8 waves/block, 16 edges per wave; weights staged once per block in LDS
// (zero-padded + pair-swizzled); all three GEMM layers on f32 WMMA.
//   h1 = gelu(edge_attr @ W1 + b1)   (16x17 @ 17x48, K padded to 20)
//   h2 = relu(h1 @ W2 + b2)          (16x48 @ 48x48)
//   g  = gelu(h2 @ Wg + bg)          (16x48 @ 48x48)
//   w_e = g . Ww + bw
// ---------------------------------------------------------------------------
__global__ __launch_bounds__(256) void edge_mlp_kernel(
        const float* __restrict__ edge_attr,
        const float* __restrict__ W1, const float* __restrict__ b1,
        const float* __restrict__ W2, const float* __restrict__ b2,
        const float* __restrict__ Wg, const float* __restrict__ bg,
        const float* __restrict__ Ww, const float* __restrict__ bw,
        float* __restrict__ w_e) {
    __shared__ __align__(16) float Wp1[20 * 48];               // zero-padded rows 17..19
    __shared__ __align__(16) float Wp2[48 * 48];
    __shared__ __align__(16) float Wpg[48 * 48];
    __shared__ __align__(16) float As[MLP_WAVES][16 * 20];     // edge_attr tiles (K pad 20)
    __shared__ __align__(16) float Hs[MLP_WAVES][16 * 48];     // inter-layer activations

    const int tid   = threadIdx.x;
    const int wave  = tid >> 5;
    const int lane  = tid & 31;
    const int row   = lane & 15;
    const int khalf = lane >> 4;
    const int e0    = blockIdx.x * MLP_EPB + wave * 16;

    // ---- Stage pair-swizzled weights (once per 128 edges) ----
    for (int idx = tid; idx < 20 * 48; idx += 256) {
        int k = idx / 48, c = idx - k * 48;
        Wp1[(k >> 1) * 96 + c * 2 + (k & 1)] = (k < EDD) ? W1[idx] : 0.0f;
    }
    for (int idx = tid; idx < 48 * 48; idx += 256) {
        int k = idx / 48, c = idx - k * 48;
        int sw = (k >> 1) * 96 + c * 2 + (k & 1);
        Wp2[sw] = W2[idx];
        Wpg[sw] = Wg[idx];
    }
    // ---- Stage this wave's 16x17 edge_attr tile (zero-padded to K=20) ----
    float* Aw = As[wave];
    for (int idx = lane; idx < 16 * 20; idx += 32) {
        int r = idx / 20, k = idx - r * 20;
        Aw[idx] = (k < EDD) ? edge_attr[(size_t)(e0 + r) * EDD + k] : 0.0f;
    }
    __syncthreads();

    float* Hw = Hs[wave];
    const v8f z = {};
    v8f acc[3];

    // ---- Layer 1: 16x20(pad) @ 20x48 ----
    acc[0] = z; acc[1] = z; acc[2] = z;
    gemm_tile_lds<5, 20>(Aw, Wp1, row, khalf, acc);
    __syncthreads();
#pragma unroll
    for (int nb = 0; nb < 3; ++nb) {
        const int n = nb * 16 + row;
        const float bias = b1[n];
#pragma unroll
        for (int r = 0; r < 8; ++r)
            Hw[(r + khalf * 8) * 48 + n] = gelu_exact(acc[nb][r] + bias);
    }
    __syncthreads();

    // ---- Layer 2: 16x48 @ 48x48 ----
    acc[0] = z; acc[1] = z; acc[2] = z;
    gemm_tile_lds<12, 48>(Hw, Wp2, row, khalf, acc);
    __syncthreads();
#pragma unroll
    for (int nb = 0; nb < 3; ++nb) {
        const int n = nb * 16 + row;
        const float bias = b2[n];
#pragma unroll
        for (int r = 0; r < 8; ++r) {
            float h = acc[nb][r] + bias;
            Hw[(r + khalf * 8) * 48 + n] = (h > 0.0f) ? h : 0.0f;
        }
    }
    __syncthreads();

    // ---- Layer 3: 16x48 @ 48x48 ----
    acc[0] = z; acc[1] = z; acc[2] = z;
    gemm_tile_lds<12, 48>(Hw, Wpg, row, khalf, acc);
    __syncthreads();
#pragma unroll
    for (int nb = 0; nb < 3; ++nb) {
        const int n = nb * 16 + row;
        const float bias = bg[n];
#pragma unroll
        for (int r = 0; r < 8; ++r)
            Hw[(r + khalf * 8) * 48 + n] = gelu_exact(acc[nb][r] + bias);
    }
    __syncthreads();

    // ---- Final dot with Ww: one edge per lane 0..15 of each wave ----
    if (lane < 16) {
        float w = bw[0];
#pragma unroll
        for (int n = 0; n < HH; ++n) w += Hw[lane * 48 + n] * Ww[n];
        w_e[e0 + lane] = w;
    }
}

// ---------------------------------------------------------------------------
// Kernel: per-edge gated scatter-add. Half-wave (16 lanes) per edge.
//   s = sigmoid(vmean[src] * w_e[e]);  agg[dst, 0:48] += x[src, 0:48] * s
// (columns L:2L of agg are never consumed by the reference -> skipped)
// ---------------------------------------------------------------------------
__global__ void edge_pass_kernel(const int* __restrict__ src, const int* __restrict__ dst,
                                 const float* __restrict__ w_e,
                                 const float* __restrict__ x, const float* __restrict__ vmean,
                                 float* __restrict__ agg) {
    int t = blockIdx.x * blockDim.x + threadIdx.x;
    int e = t >> 4;
    int j = t & 15;
    if (e >= EE) return;
    int s = src[e];
    int d = dst[e];
    float sg = sigmoidf_(vmean[s] * w_e[e]);
    const float* xs = x + (size_t)s * LL;
    float* ad = agg + (size_t)d * LL;
    atomicAdd(&ad[j],      xs[j]      * sg);
    atomicAdd(&ad[j + 16], xs[j + 16] * sg);
    atomicAdd(&ad[j + 32], xs[j + 32] * sg);
}

// ---------------------------------------------------------------------------
// Kernel: node update — gated blend, validity propagation, agg re-zero
// ---------------------------------------------------------------------------
__global__ void node_update_kernel(float* __restrict__ x, float* __restrict__ v,
                                   const float* __restrict__ orig, float* __restrict__ agg,
                                   const float* __restrict__ Wf, const float* __restrict__ bf,
                                   int n) {
    int t = blockIdx.x * blockDim.x + threadIdx.x;
    if (t >= n) return;
    int l = t % LL;
    float nh = agg[t];
    float xo = x[t];
    float vv = v[t];
    float nv = 1.0f - vv;
    float m = sigmoidf_(Wf[0] * nh + Wf[1] * xo + Wf[2] * nv + bf[0]);
    float xn = (1.0f - m) * xo + nv * m * nh;
    float vn = ((orig[t] != xn) || (vv > 0.0f)) ? 1.0f : 0.0f;
    if (l == 0) vn = 0.0f;
    x[t] = xn;
    v[t] = vn;
    agg[t] = 0.0f;   // re-zero for next iteration
}

// ---------------------------------------------------------------------------
// Kernel: emit x[0][:48]
// ---------------------------------------------------------------------------
__global__ void out_kernel(const float* __restrict__ x, float* __restrict__ out) {
    int t = threadIdx.x;
    if (t < LL) out[t] = x[t];
}

// ---------------------------------------------------------------------------
extern "C" void kernel_launch(void* const* d_in, const int* in_sizes, int n_in,
                              void* d_out, int out_size, void* d_ws, size_t ws_size,
                              hipStream_t stream) {
    const float* nodes     = (const float*)d_in[0];
    const int*   eidx      = (const int*)  d_in[1];
    const float* edge_attr = (const float*)d_in[2];
    const float* valid     = (const float*)d_in[3];
    const float* W1 = (const float*)d_in[6];
    const float* b1 = (const float*)d_in[7];
    const float* W2 = (const float*)d_in[8];
    const float* b2 = (const float*)d_in[9];
    const float* Wg = (const float*)d_in[10];
    const float* bg = (const float*)d_in[11];
    const float* Ww = (const float*)d_in[12];
    const float* bw = (const float*)d_in[13];
    const float* Wf = (const float*)d_in[14];
    const float* bf = (const float*)d_in[15];

    const size_t NL = (size_t)NN * LL;
    float* ws    = (float*)d_ws;
    float* x     = ws;                 // N*L
    float* v     = ws + NL;            // N*L
    float* orig  = ws + 2 * NL;        // N*L
    float* agg   = ws + 3 * NL;        // N*L
    float* w_e   = ws + 4 * NL;        // E
    float* vmean = ws + 4 * NL + EE;   // N
    // total workspace: 4*N*L + E + N floats ~= 83.6 MB

    const int* src = eidx;
    const int* dst = eidx + EE;

    const int nNL = (int)NL;
    init_kernel<<<(nNL + 255) / 256, 256, 0, stream>>>(nodes, valid, x, v, orig, agg, nNL);

    // Loop-invariant edge MLP (once): 87 v_wmma_f32_16x16x4_f32 per wave
    edge_mlp_kernel<<<EE / MLP_EPB, 256, 0, stream>>>(edge_attr, W1, b1, W2, b2, Wg, bg, Ww, bw, w_e);

    for (int it = 0; it < 3; ++it) {
        vmean_kernel<<<(NN * 16 + 255) / 256, 256, 0, stream>>>(v, vmean, NN);
        edge_pass_kernel<<<(EE * 16 + 255) / 256, 256, 0, stream>>>(src, dst, w_e, x, vmean, agg);
        node_update_kernel<<<(nNL + 255) / 256, 256, 0, stream>>>(x, v, orig, agg, Wf, bf, nNL);
    }

    out_kernel<<<1, 64, 0, stream>>>(x, (float*)d_out);
}